// GCNConnectivity_7146825580734
// MI455X (gfx1250) — compile-verified
//
#include <hip/hip_runtime.h>
#include <math.h>

#define GN 8192   // nodes
#define GE 524288 // edges (without self loops)
#define GF 512    // input features
#define GH 64     // hidden

typedef __attribute__((ext_vector_type(2))) float v2f;
typedef __attribute__((ext_vector_type(8))) float v8f;

// ---------------------------------------------------------------------------
// Generic f32 WMMA GEMM: C[M,Nc] = A[M,K] @ B[K,Nc], all row-major.
// One wave32 per 16x16 output tile, accumulating with V_WMMA_F32_16X16X4_F32.
// grid = (Nc/16, M/16), block = 32.
// ---------------------------------------------------------------------------
__global__ __launch_bounds__(32) void k_gemm_wmma(
    const float* __restrict__ A, const float* __restrict__ B,
    float* __restrict__ C, int K, int Nc) {
  const int tileN = blockIdx.x;
  const int tileM = blockIdx.y;
  const int lane  = threadIdx.x;
  const int half  = lane >> 4;   // 0: K pair {0,1}; 1: K pair {2,3}
  const int lid   = lane & 15;   // row (A) / col (B)

  v8f acc = {};
  const float* Arow = A + (size_t)(tileM * 16 + lid) * K + 2 * half;
  const float* Bcol = B + (size_t)(2 * half) * Nc + tileN * 16 + lid;

  for (int k = 0; k < K; k += 4) {
    v2f a, b;
    a.x = Arow[k + 0];
    a.y = Arow[k + 1];
    b.x = Bcol[(size_t)k * Nc];
    b.y = Bcol[(size_t)k * Nc + Nc];
    acc = __builtin_amdgcn_wmma_f32_16x16x4_f32(
        /*neg_a=*/false, a, /*neg_b=*/false, b,
        /*c_mod=*/(short)0, acc, /*reuse_a=*/false, /*reuse_b=*/false);
  }

  // D layout: VGPR r -> row (r + 8*half), col lid
  float* Crow = C + (size_t)(tileM * 16 + 8 * half) * Nc + tileN * 16 + lid;
#pragma unroll
  for (int r = 0; r < 8; ++r) Crow[(size_t)r * Nc] = acc[r];
}

// ---------------------------------------------------------------------------
// Final fused kernel: out = 0.5*(T + T^T), T = tanh(h2 @ Wfc + bfc).
// Each block (J >= I) computes tile (I,J) and tile (J,I) with two WMMA
// accumulator chains, combines via an LDS transpose, writes both tiles.
// grid = (N/16, N/16), block = 32.
// ---------------------------------------------------------------------------
__global__ __launch_bounds__(32) void k_out_sym(
    const float* __restrict__ h2, const float* __restrict__ Wfc,
    const float* __restrict__ bfc, float* __restrict__ out) {
  const int J = blockIdx.x;
  const int I = blockIdx.y;
  if (J < I) return;  // uniform per-block: EXEC stays all-1s for WMMA blocks

  const int lane = threadIdx.x;
  const int half = lane >> 4;
  const int lid  = lane & 15;

  __shared__ float tile[16][17];

  v8f acc1 = {};  // T1 = h2[I-rows] @ Wfc[:, J-cols]
  v8f acc2 = {};  // T2 = h2[J-rows] @ Wfc[:, I-cols]
  const float* A1 = h2 + (size_t)(I * 16 + lid) * GH + 2 * half;
  const float* A2 = h2 + (size_t)(J * 16 + lid) * GH + 2 * half;
  const float* B1 = Wfc + (size_t)(2 * half) * GN + J * 16 + lid;
  const float* B2 = Wfc + (size_t)(2 * half) * GN + I * 16 + lid;

#pragma unroll
  for (int k = 0; k < GH; k += 4) {
    v2f a1, a2, b1, b2;
    a1.x = A1[k];  a1.y = A1[k + 1];
    a2.x = A2[k];  a2.y = A2[k + 1];
    b1.x = B1[(size_t)k * GN];  b1.y = B1[(size_t)k * GN + GN];
    b2.x = B2[(size_t)k * GN];  b2.y = B2[(size_t)k * GN + GN];
    acc1 = __builtin_amdgcn_wmma_f32_16x16x4_f32(false, a1, false, b1,
                                                 (short)0, acc1, false, false);
    acc2 = __builtin_amdgcn_wmma_f32_16x16x4_f32(false, a2, false, b2,
                                                 (short)0, acc2, false, false);
  }

  const float bJ = bfc[J * 16 + lid];
  const float bI = bfc[I * 16 + lid];
#pragma unroll
  for (int r = 0; r < 8; ++r) {
    acc1[r] = tanhf(acc1[r] + bJ);
    acc2[r] = tanhf(acc2[r] + bI);
  }

  // S = 0.5*(T1 + T2^T): transpose T2 through LDS
#pragma unroll
  for (int r = 0; r < 8; ++r) tile[r + 8 * half][lid] = acc2[r];
  __syncthreads();
  v8f s;
#pragma unroll
  for (int r = 0; r < 8; ++r)
    s[r] = 0.5f * (acc1[r] + tile[lid][r + 8 * half]);

  // out[I-tile, J-tile] = S
  float* O1 = out + (size_t)(I * 16 + 8 * half) * GN + J * 16 + lid;
#pragma unroll
  for (int r = 0; r < 8; ++r) O1[(size_t)r * GN] = s[r];

  if (J != I) {
    // out[J-tile, I-tile] = S^T: transpose S through LDS
    __syncthreads();
#pragma unroll
    for (int r = 0; r < 8; ++r) tile[r + 8 * half][lid] = s[r];
    __syncthreads();
    float* O2 = out + (size_t)(J * 16 + 8 * half) * GN + I * 16 + lid;
#pragma unroll
    for (int r = 0; r < 8; ++r) O2[(size_t)r * GN] = tile[lid][r + 8 * half];
  }
}

// ---------------------------------------------------------------------------
// Graph-aggregation helpers
// ---------------------------------------------------------------------------
__global__ void k_fill(float* __restrict__ p, float v, int n) {
  int i = blockIdx.x * blockDim.x + threadIdx.x;
  if (i < n) p[i] = v;
}

// deg += 1 per incoming edge (deg pre-initialized to 1.0 for the self loop)
__global__ void k_deg(const long long* __restrict__ ei, float* __restrict__ deg) {
  int e = blockIdx.x * blockDim.x + threadIdx.x;
  if (e >= GE) return;
  int d = (int)ei[GE + e];  // edge_index[1][e]
  atomicAdd(&deg[d], 1.0f);
}

__global__ void k_dinv(float* __restrict__ p) {
  int i = blockIdx.x * blockDim.x + threadIdx.x;
  if (i >= GN) return;
  float d = p[i];
  p[i] = (d > 0.0f) ? rsqrtf(d) : 0.0f;
}

// hout[i*H + f] = b[f]  (bias pre-load before scatter-add)
__global__ void k_bias_init(float* __restrict__ h, const float* __restrict__ b) {
  int i = blockIdx.x * blockDim.x + threadIdx.x;  // exact multiple of GN*GH
  h[i] = b[i & (GH - 1)];
}

// messages: hout[dst] += hin[src] * dinv[src] * dinv[dst]  (E real edges + N self loops)
__global__ void k_scatter(const float* __restrict__ hin,
                          const long long* __restrict__ ei,
                          const float* __restrict__ dinv,
                          float* __restrict__ hout) {
  long long t = (long long)blockIdx.x * blockDim.x + threadIdx.x;
  int e = (int)(t >> 6);   // /GH
  int f = (int)(t & 63);   // %GH
  if (e >= GE + GN) return;
  int s, d;
  if (e < GE) {
    s = (int)ei[e];        // edge_index[0][e]
    d = (int)ei[GE + e];   // edge_index[1][e]
  } else {
    s = d = e - GE;        // self loop
  }
  float v = hin[(size_t)s * GH + f] * dinv[s] * dinv[d];
  atomicAdd(&hout[(size_t)d * GH + f], v);
}

__global__ void k_relu(float* __restrict__ h) {
  int i = blockIdx.x * blockDim.x + threadIdx.x;  // exact multiple of GN*GH
  h[i] = fmaxf(h[i], 0.0f);
}

// ---------------------------------------------------------------------------
extern "C" void kernel_launch(void* const* d_in, const int* in_sizes, int n_in,
                              void* d_out, int out_size, void* d_ws, size_t ws_size,
                              hipStream_t stream) {
  const float*     x   = (const float*)d_in[0];
  const long long* ei  = (const long long*)d_in[1];  // int64 per reference
  const float*     W1  = (const float*)d_in[2];
  const float*     b1  = (const float*)d_in[3];
  const float*     W2  = (const float*)d_in[4];
  const float*     b2  = (const float*)d_in[5];
  const float*     Wfc = (const float*)d_in[6];
  const float*     bfc = (const float*)d_in[7];
  float* out = (float*)d_out;

  // workspace: dinv [N] | bufA [N*H] | bufB [N*H]   (~4.2 MB total)
  float* dinv = (float*)d_ws;
  float* bufA = dinv + GN;
  float* bufB = bufA + (size_t)GN * GH;

  const int NH = GN * GH;  // 524288
  const long long scat_threads = (long long)(GE + GN) * GH;
  const unsigned scat_blocks = (unsigned)((scat_threads + 255) / 256);

  // degree / normalization
  k_fill<<<GN / 256, 256, 0, stream>>>(dinv, 1.0f, GN);   // self loop
  k_deg<<<GE / 256, 256, 0, stream>>>(ei, dinv);
  k_dinv<<<GN / 256, 256, 0, stream>>>(dinv);

  // ---- GCNConv 1: bufB = relu(scatter(x@W1) + b1) ----
  k_gemm_wmma<<<dim3(GH / 16, GN / 16), 32, 0, stream>>>(x, W1, bufA, GF, GH);
  k_bias_init<<<NH / 256, 256, 0, stream>>>(bufB, b1);
  k_scatter<<<scat_blocks, 256, 0, stream>>>(bufA, ei, dinv, bufB);
  k_relu<<<NH / 256, 256, 0, stream>>>(bufB);

  // ---- GCNConv 2: bufB = relu(scatter(bufB@W2) + b2) ----
  k_gemm_wmma<<<dim3(GH / 16, GN / 16), 32, 0, stream>>>(bufB, W2, bufA, GH, GH);
  k_bias_init<<<NH / 256, 256, 0, stream>>>(bufB, b2);
  k_scatter<<<scat_blocks, 256, 0, stream>>>(bufA, ei, dinv, bufB);
  k_relu<<<NH / 256, 256, 0, stream>>>(bufB);

  // ---- fused final: out = 0.5*(tanh(bufB@Wfc + bfc) + transpose) ----
  k_out_sym<<<dim3(GN / 16, GN / 16), 32, 0, stream>>>(bufB, Wfc, bfc, out);
}